// HSTUBlockPreprocessor_87763361726750
// MI455X (gfx1250) — compile-verified
//
#include <hip/hip_runtime.h>
#include <stdint.h>

// ---------------------------------------------------------------------------
// HSTU block preprocessor: jagged scatter-concat.
// Pure data movement (0 FLOPs, ~203 MB traffic -> ~8.7us at 23.3 TB/s HBM).
// Strategy: one wave32 per 512-float output row; scalar binary search gives
// the destination row; row is moved via the CDNA5 async data mover
// (global_load_async_to_lds_b128 / global_store_async_from_lds_b128) with
// s_wait_asynccnt fences. 16B/lane granularity, one read + one write per byte.
// ---------------------------------------------------------------------------

#define DD 512                 // embedding dim (floats per row)
#define WAVES_PER_BLOCK 8
#define THREADS (WAVES_PER_BLOCK * 32)

typedef int v4i __attribute__((vector_size(16)));

// ---- CDNA5 async global<->LDS helpers -------------------------------------
#if defined(__has_builtin)
#if __has_builtin(__builtin_amdgcn_global_load_async_to_lds_b128)
#define HAVE_ASYNC_BUILTIN 1
#endif
#if __has_builtin(__builtin_amdgcn_s_wait_asynccnt)
#define HAVE_ASYNCCNT_BUILTIN 1
#endif
#endif

__device__ __forceinline__ void async_g2l_b128(const float* g, float* l) {
#ifdef HAVE_ASYNC_BUILTIN
  __builtin_amdgcn_global_load_async_to_lds_b128(
      (__attribute__((address_space(1))) v4i*)g,
      (__attribute__((address_space(3))) v4i*)l, 0, 0);
#else
  unsigned l32 = (unsigned)(unsigned long)((__attribute__((address_space(3))) char*)l);
  asm volatile("global_load_async_to_lds_b128 %0, %1, off"
               :: "v"(l32), "v"(g) : "memory");
#endif
}

__device__ __forceinline__ void async_l2g_b128(float* g, const float* l) {
#ifdef HAVE_ASYNC_BUILTIN
  __builtin_amdgcn_global_store_async_from_lds_b128(
      (__attribute__((address_space(1))) v4i*)g,
      (__attribute__((address_space(3))) v4i*)l, 0, 0);
#else
  unsigned l32 = (unsigned)(unsigned long)((__attribute__((address_space(3))) char*)l);
  asm volatile("global_store_async_from_lds_b128 %0, %1, off"
               :: "v"(g), "v"(l32) : "memory");
#endif
}

__device__ __forceinline__ void wait_asynccnt0() {
#ifdef HAVE_ASYNCCNT_BUILTIN
  __builtin_amdgcn_s_wait_asynccnt(0);
#else
  asm volatile("s_wait_asynccnt 0" ::: "memory");
#endif
}

// find b in [0, nb) such that off[b] <= i < off[b+1]; off has nb+1 entries.
// Wave-uniform -> scalarized by the compiler; offsets arrays are 516B (cached).
__device__ __forceinline__ int find_bucket(const int* __restrict__ off, int nb, int i) {
  int lo = 0, hi = nb;
  while (lo + 1 < hi) {
    int mid = (lo + hi) >> 1;
    if (off[mid] <= i) lo = mid; else hi = mid;
  }
  return lo;
}

__global__ __launch_bounds__(THREADS)
void hstu_scatter_kernel(const float* __restrict__ item, const float* __restrict__ action,
                         const float* __restrict__ ctx1v, const float* __restrict__ ctx2v,
                         const int* __restrict__ io, const int* __restrict__ c1,
                         const int* __restrict__ c2, float* __restrict__ out,
                         int n_item, int n_ctx1, int n_ctx2, int nb, int total_rows) {
  __shared__ __attribute__((aligned(16))) float smem[WAVES_PER_BLOCK * DD];
  const int wave = threadIdx.x >> 5;
  const int lane = threadIdx.x & 31;
  const int row  = blockIdx.x * WAVES_PER_BLOCK + wave;
  if (row >= total_rows) return;   // uniform per wave

  const float* src;
  long dst_row;
  if (row < n_item) {                       // item row i -> even interleave slot
    int i = row;
    int b = find_bucket(io, nb, i);
    dst_row = (long)c1[b + 1] + (long)c2[b + 1] + 2L * i;
    src = item + (size_t)i * DD;
  } else if (row < 2 * n_item) {            // action row i -> odd interleave slot
    int i = row - n_item;
    int b = find_bucket(io, nb, i);
    dst_row = (long)c1[b + 1] + (long)c2[b + 1] + 2L * i + 1L;
    src = action + (size_t)i * DD;
  } else if (row < 2 * n_item + n_ctx1) {   // ctx1 prefix rows
    int i = row - 2 * n_item;
    int b = find_bucket(c1, nb, i);
    dst_row = (long)c2[b] + 2L * io[b] + i;
    src = ctx1v + (size_t)i * DD;
  } else {                                  // ctx2 prefix rows (after ctx1 of same sample)
    int i = row - 2 * n_item - n_ctx1;
    int b = find_bucket(c2, nb, i);
    dst_row = (long)c1[b + 1] + 2L * io[b] + i;
    src = ctx2v + (size_t)i * DD;
  }

  float* dst  = out + (size_t)dst_row * DD;
  float* lbuf = smem + wave * DD;

  // 2KB row: 4 async B128 transfers per direction (32 lanes x 16B = 512B each).
#pragma unroll
  for (int c = 0; c < 4; ++c)
    async_g2l_b128(src + c * 128 + lane * 4, lbuf + c * 128 + lane * 4);
  wait_asynccnt0();
#pragma unroll
  for (int c = 0; c < 4; ++c)
    async_l2g_b128(dst + c * 128 + lane * 4, lbuf + c * 128 + lane * 4);
  wait_asynccnt0();
}

// merged_lengths[b] = (c1[b+1]-c1[b]) + (c2[b+1]-c2[b]) + 2*(io[b+1]-io[b]),
// written as float into the tail of d_out (second tuple output, flattened).
__global__ void hstu_lengths_kernel(const int* __restrict__ io, const int* __restrict__ c1,
                                    const int* __restrict__ c2, float* __restrict__ out_tail,
                                    int nb) {
  int b = blockIdx.x * blockDim.x + threadIdx.x;
  if (b < nb) {
    int len = (c1[b + 1] - c1[b]) + (c2[b + 1] - c2[b]) + 2 * (io[b + 1] - io[b]);
    out_tail[b] = (float)len;
  }
}

extern "C" void kernel_launch(void* const* d_in, const int* in_sizes, int n_in,
                              void* d_out, int out_size, void* d_ws, size_t ws_size,
                              hipStream_t stream) {
  (void)n_in; (void)out_size; (void)d_ws; (void)ws_size;
  const float* item   = (const float*)d_in[0];
  const float* action = (const float*)d_in[1];
  const float* ctx1v  = (const float*)d_in[2];
  const float* ctx2v  = (const float*)d_in[3];
  const int*   io     = (const int*)d_in[4];
  const int*   c1     = (const int*)d_in[5];
  const int*   c2     = (const int*)d_in[6];

  const int n_item = in_sizes[0] / DD;
  const int n_ctx1 = in_sizes[2] / DD;
  const int n_ctx2 = in_sizes[3] / DD;
  const int nb     = in_sizes[4] - 1;
  const int total_rows = 2 * n_item + n_ctx1 + n_ctx2;

  float* out = (float*)d_out;
  const int grid = (total_rows + WAVES_PER_BLOCK - 1) / WAVES_PER_BLOCK;
  hstu_scatter_kernel<<<grid, THREADS, 0, stream>>>(item, action, ctx1v, ctx2v,
                                                    io, c1, c2, out,
                                                    n_item, n_ctx1, n_ctx2, nb, total_rows);
  hstu_lengths_kernel<<<1, 256, 0, stream>>>(io, c1, c2,
                                             out + (size_t)total_rows * DD, nb);
}